// Server_Centroids_GNN_6631429505543
// MI455X (gfx1250) — compile-verified
//
#include <hip/hip_runtime.h>
#include <hip/hip_bf16.h>

typedef float v2f __attribute__((ext_vector_type(2)));
typedef float v8f __attribute__((ext_vector_type(8)));

#define NND 4096
#define DD  256
#define K_SEL 1677721u   // int(4096*4096*0.1)

// ---------------- WMMA f32 16x16x4 wrapper ----------------
__device__ __forceinline__ v8f wmma4(v2f a, v2f b, v8f c) {
  return __builtin_amdgcn_wmma_f32_16x16x4_f32(false, a, false, b, (short)0, c,
                                               false, false);
}

// ---------------- block reduction (256 threads) ----------------
__device__ __forceinline__ float block_sum(float v, float* red) {
  const int tid = threadIdx.x;
  red[tid] = v;
  __syncthreads();
  for (int s = 128; s > 0; s >>= 1) {
    if (tid < s) red[tid] += red[tid + s];
    __syncthreads();
  }
  float r = red[0];
  __syncthreads();
  return r;
}

// ---------------- LayerNorm + cosine row-normalize ----------------
__global__ __launch_bounds__(256) void ln_norm_kernel(
    const float* __restrict__ x, const float* __restrict__ g,
    const float* __restrict__ b, float* __restrict__ yln,
    float* __restrict__ yn) {
  __shared__ float red[256];
  const int row = blockIdx.x, tid = threadIdx.x;
  float v = x[(size_t)row * DD + tid];
  float mean = block_sum(v, red) * (1.0f / DD);
  float d = v - mean;
  float var = block_sum(d * d, red) * (1.0f / DD);
  float y = d * rsqrtf(var + 1e-5f) * g[tid] + b[tid];
  yln[(size_t)row * DD + tid] = y;
  float ss = block_sum(y * y, red);
  float nrm = fmaxf(sqrtf(ss), 1e-8f);
  yn[(size_t)row * DD + tid] = y / nrm;
}

// ---- residual + LayerNorm + relu + cosine row-normalize (layer1 out) ----
__global__ __launch_bounds__(256) void resid_ln_relu_norm_kernel(
    const float* __restrict__ t, const float* __restrict__ res,
    const float* __restrict__ g, const float* __restrict__ b,
    float* __restrict__ h, float* __restrict__ hn) {
  __shared__ float red[256];
  const int row = blockIdx.x, tid = threadIdx.x;
  float v = t[(size_t)row * DD + tid] + res[(size_t)row * DD + tid];
  float mean = block_sum(v, red) * (1.0f / DD);
  float d = v - mean;
  float var = block_sum(d * d, red) * (1.0f / DD);
  float y = d * rsqrtf(var + 1e-5f) * g[tid] + b[tid];
  y = fmaxf(y, 0.0f);
  h[(size_t)row * DD + tid] = y;
  float ss = block_sum(y * y, red);
  float nrm = fmaxf(sqrtf(ss), 1e-8f);
  hn[(size_t)row * DD + tid] = y / nrm;
}

// ---------------- S = P * P^T  (4096x256 -> 4096x4096), 64x64 tile/wave ----
__global__ __launch_bounds__(128) void gemm_nt_sim_kernel(
    const float* __restrict__ P, float* __restrict__ S) {
  const int wave = threadIdx.x >> 5;
  const int lane = threadIdx.x & 31;
  const int half = lane >> 4;
  const int l = lane & 15;
  const int t = blockIdx.x * 4 + wave;
  const int row0 = (t >> 6) * 64;
  const int col0 = (t & 63) * 64;
  const float* Ap = P + (size_t)(row0 + l) * DD + 2 * half;
  const float* Bp = P + (size_t)(col0 + l) * DD + 2 * half;
  const v8f z = {0.f, 0.f, 0.f, 0.f, 0.f, 0.f, 0.f, 0.f};
  v8f acc[4][4];
#pragma unroll
  for (int m = 0; m < 4; ++m)
#pragma unroll
    for (int n = 0; n < 4; ++n) acc[m][n] = z;

  for (int k = 0; k < DD; k += 4) {
    v2f a[4], b[4];
#pragma unroll
    for (int m = 0; m < 4; ++m)
      a[m] = *(const v2f*)(Ap + (size_t)(16 * m) * DD + k);
#pragma unroll
    for (int n = 0; n < 4; ++n)
      b[n] = *(const v2f*)(Bp + (size_t)(16 * n) * DD + k);
#pragma unroll
    for (int m = 0; m < 4; ++m)
#pragma unroll
      for (int n = 0; n < 4; ++n) acc[m][n] = wmma4(a[m], b[n], acc[m][n]);
  }
#pragma unroll
  for (int m = 0; m < 4; ++m)
#pragma unroll
    for (int n = 0; n < 4; ++n)
#pragma unroll
      for (int r = 0; r < 8; ++r)
        S[(size_t)(row0 + 16 * m + 8 * half + r) * NND + col0 + 16 * n + l] =
            acc[m][n][r];
}

// ------- C[4096x256] = A[4096xK] * B[Kx256] (+resid), 32x32 tile/wave -------
__global__ __launch_bounds__(256) void gemm_nn_kernel(
    const float* __restrict__ A, int K, const float* __restrict__ B,
    const float* __restrict__ resid, float* __restrict__ C) {
  const int wave = threadIdx.x >> 5;
  const int lane = threadIdx.x & 31;
  const int half = lane >> 4;
  const int l = lane & 15;
  const int t = blockIdx.x * 8 + wave;
  const int row0 = (t >> 3) * 32;
  const int col0 = (t & 7) * 32;
  const float* Ap = A + (size_t)(row0 + l) * K + 2 * half;
  const v8f z = {0.f, 0.f, 0.f, 0.f, 0.f, 0.f, 0.f, 0.f};
  v8f a00 = z, a01 = z, a10 = z, a11 = z;

  for (int k = 0; k < K; k += 4) {
    v2f a0 = *(const v2f*)(Ap + k);
    v2f a1 = *(const v2f*)(Ap + (size_t)16 * K + k);
    const float* Bp = B + (size_t)(k + 2 * half) * DD + col0 + l;
    v2f b0, b1;
    b0.x = Bp[0];
    b0.y = Bp[DD];
    b1.x = Bp[16];
    b1.y = Bp[16 + DD];
    a00 = wmma4(a0, b0, a00);
    a01 = wmma4(a0, b1, a01);
    a10 = wmma4(a1, b0, a10);
    a11 = wmma4(a1, b1, a11);
  }
  v8f acc[2][2] = {{a00, a01}, {a10, a11}};
#pragma unroll
  for (int m = 0; m < 2; ++m)
#pragma unroll
    for (int n = 0; n < 2; ++n)
#pragma unroll
      for (int r = 0; r < 8; ++r) {
        size_t off =
            (size_t)(row0 + 16 * m + 8 * half + r) * DD + col0 + 16 * n + l;
        float v = acc[m][n][r];
        if (resid) v += resid[off];
        C[off] = v;
      }
}

// ---------------- exact top-k threshold: radix-select ----------------
__device__ __forceinline__ unsigned fkey(float f) {
  unsigned u = __float_as_uint(f);
  return (u & 0x80000000u) ? ~u : (u | 0x80000000u);
}

__global__ void hist_hi_kernel(const float* __restrict__ sim,
                               unsigned* __restrict__ hist) {
  int i = blockIdx.x * blockDim.x + threadIdx.x;
  const int stride = gridDim.x * blockDim.x;
  for (; i < NND * NND; i += stride) {
    unsigned kv = fkey(sim[i]);
    atomicAdd(&hist[kv >> 16], 1u);
  }
}

__global__ void hist_lo_kernel(const float* __restrict__ sim,
                               const unsigned* __restrict__ sc,
                               unsigned* __restrict__ hist) {
  const unsigned b1 = sc[0];
  int i = blockIdx.x * blockDim.x + threadIdx.x;
  const int stride = gridDim.x * blockDim.x;
  for (; i < NND * NND; i += stride) {
    unsigned kv = fkey(sim[i]);
    if ((kv >> 16) == b1) atomicAdd(&hist[kv & 0xFFFFu], 1u);
  }
}

// pass 0: find hi-bin + residual rank;  writes sc[0]=bin, sc[1]=rank
__global__ __launch_bounds__(256) void select_hi_kernel(
    const unsigned* __restrict__ hist, unsigned* __restrict__ sc) {
  __shared__ unsigned csum[256];
  const int tid = threadIdx.x;
  const int base = (255 - tid) << 8;  // chunks in descending value order
  unsigned s = 0;
  for (int j = 0; j < 256; ++j) s += hist[base + j];
  csum[tid] = s;
  __syncthreads();
  if (tid == 0) {
    const unsigned k = K_SEL;
    unsigned cum = 0, before = 0;
    int sel = 255;
    for (int c = 0; c < 256; ++c) {
      if (cum + csum[c] >= k) { sel = c; before = cum; break; }
      cum += csum[c];
    }
    const int cb = (255 - sel) << 8;
    unsigned rank = k - before;
    unsigned cum2 = 0;
    int bin = cb;
    for (int j = 255; j >= 0; --j) {
      unsigned h = hist[cb + j];
      if (cum2 + h >= rank) { bin = cb + j; rank -= cum2; break; }
      cum2 += h;
    }
    sc[0] = (unsigned)bin;
    sc[1] = rank;
  }
}

// pass 1: find lo 16 bits, reconstruct threshold float into ((float*)sc)[2]
__global__ __launch_bounds__(256) void select_lo_kernel(
    const unsigned* __restrict__ hist, unsigned* __restrict__ sc) {
  __shared__ unsigned csum[256];
  const int tid = threadIdx.x;
  const int base = (255 - tid) << 8;
  unsigned s = 0;
  for (int j = 0; j < 256; ++j) s += hist[base + j];
  csum[tid] = s;
  __syncthreads();
  if (tid == 0) {
    const unsigned k = sc[1];
    unsigned cum = 0, before = 0;
    int sel = 255;
    for (int c = 0; c < 256; ++c) {
      if (cum + csum[c] >= k) { sel = c; before = cum; break; }
      cum += csum[c];
    }
    const int cb = (255 - sel) << 8;
    unsigned rank = k - before;
    unsigned cum2 = 0;
    int bin = cb;
    for (int j = 255; j >= 0; --j) {
      unsigned h = hist[cb + j];
      if (cum2 + h >= rank) { bin = cb + j; break; }
      cum2 += h;
    }
    unsigned key = (sc[0] << 16) | (unsigned)bin;
    unsigned bits = (key & 0x80000000u) ? (key & 0x7FFFFFFFu) : ~key;
    ((float*)sc)[2] = __uint_as_float(bits);
  }
}

// ------- m = relu(thresholded sim) + I (in place); deg = rsqrt(rowsum) ------
__global__ __launch_bounds__(256) void build_m_deg_kernel(
    float* __restrict__ sim, float* __restrict__ deg,
    const float* __restrict__ scf) {
  __shared__ float red[256];
  const int row = blockIdx.x, tid = threadIdx.x;
  const float thr = scf[2] - 1e-6f;  // thr - EPS
  float lsum = 0.f;
  float* rp = sim + (size_t)row * NND;
  for (int j = tid; j < NND; j += 256) {
    float s = rp[j];
    float v = (s > thr) ? s : 0.f;
    v = fmaxf(v, 0.f);           // relu
    if (j == row) v += 1.f;      // + eye
    rp[j] = v;
    lsum += v;
  }
  float tot = block_sum(lsum, red);
  if (tid == 0) deg[row] = rsqrtf(tot);
}

// ---------------- A = D * m * D (in place) ----------------
__global__ __launch_bounds__(256) void scale_A_kernel(
    float* __restrict__ A, const float* __restrict__ deg) {
  const int idx = blockIdx.x * 256 + threadIdx.x;
  const int i = idx >> 12, j = idx & 4095;
  A[idx] *= deg[i] * deg[j];
}

// ---------------- mean pooling over 32 clients ----------------
__global__ __launch_bounds__(256) void pool_kernel(const float* __restrict__ h2,
                                                   float* __restrict__ out) {
  const int idx = blockIdx.x * 256 + threadIdx.x;  // c*256 + d
  const int c = idx >> 8, d = idx & 255;
  float s = 0.f;
  for (int cl = 0; cl < 32; ++cl)
    s += h2[((size_t)(cl * 128 + c)) * DD + d];
  out[idx] = s * (1.0f / 32.0f);
}

extern "C" void kernel_launch(void* const* d_in, const int* in_sizes, int n_in,
                              void* d_out, int out_size, void* d_ws,
                              size_t ws_size, hipStream_t stream) {
  (void)in_sizes; (void)n_in; (void)out_size; (void)ws_size;
  const float* x  = (const float*)d_in[0];
  const float* g0 = (const float*)d_in[1];
  const float* b0 = (const float*)d_in[2];
  const float* W1 = (const float*)d_in[3];
  const float* g1 = (const float*)d_in[4];
  const float* b1 = (const float*)d_in[5];
  const float* W2 = (const float*)d_in[6];
  float* out = (float*)d_out;

  char* ws = (char*)d_ws;
  float* xln = (float*)(ws + ((size_t)0 << 20));
  float* xn  = (float*)(ws + ((size_t)4 << 20));
  float* xw  = (float*)(ws + ((size_t)8 << 20));
  float* t   = (float*)(ws + ((size_t)12 << 20));
  float* h1  = (float*)(ws + ((size_t)16 << 20));
  float* h1n = (float*)(ws + ((size_t)20 << 20));
  float* deg = (float*)(ws + ((size_t)24 << 20));
  unsigned* hist = (unsigned*)(ws + ((size_t)24 << 20) + (64 << 10));
  unsigned* sc   = (unsigned*)(ws + ((size_t)24 << 20) + (512 << 10));
  float* sim = (float*)(ws + ((size_t)25 << 20));  // 64 MB

  // ---- layer 0: layernorm + cosine normalization ----
  ln_norm_kernel<<<NND, 256, 0, stream>>>(x, g0, b0, xln, xn);

  // ---- A1 ----
  gemm_nt_sim_kernel<<<1024, 128, 0, stream>>>(xn, sim);
  hipMemsetAsync(hist, 0, 65536 * sizeof(unsigned), stream);
  hist_hi_kernel<<<8192, 256, 0, stream>>>(sim, hist);
  select_hi_kernel<<<1, 256, 0, stream>>>(hist, sc);
  hipMemsetAsync(hist, 0, 65536 * sizeof(unsigned), stream);
  hist_lo_kernel<<<8192, 256, 0, stream>>>(sim, sc, hist);
  select_lo_kernel<<<1, 256, 0, stream>>>(hist, sc);
  build_m_deg_kernel<<<NND, 256, 0, stream>>>(sim, deg, (const float*)sc);
  scale_A_kernel<<<65536, 256, 0, stream>>>(sim, deg);

  // ---- layer 1: h1 = relu(LN(A1 @ (x @ W1) + x)) ----
  gemm_nn_kernel<<<128, 256, 0, stream>>>(xln, DD, W1, nullptr, xw);
  gemm_nn_kernel<<<128, 256, 0, stream>>>(sim, NND, xw, nullptr, t);
  resid_ln_relu_norm_kernel<<<NND, 256, 0, stream>>>(t, xln, g1, b1, h1, h1n);

  // ---- A2 ----
  gemm_nt_sim_kernel<<<1024, 128, 0, stream>>>(h1n, sim);
  hipMemsetAsync(hist, 0, 65536 * sizeof(unsigned), stream);
  hist_hi_kernel<<<8192, 256, 0, stream>>>(sim, hist);
  select_hi_kernel<<<1, 256, 0, stream>>>(hist, sc);
  hipMemsetAsync(hist, 0, 65536 * sizeof(unsigned), stream);
  hist_lo_kernel<<<8192, 256, 0, stream>>>(sim, sc, hist);
  select_lo_kernel<<<1, 256, 0, stream>>>(hist, sc);
  build_m_deg_kernel<<<NND, 256, 0, stream>>>(sim, deg, (const float*)sc);
  scale_A_kernel<<<65536, 256, 0, stream>>>(sim, deg);

  // ---- layer 2: h2 = A2 @ (h1 @ W2) + h1 ----
  gemm_nn_kernel<<<128, 256, 0, stream>>>(h1, DD, W2, nullptr, xw);
  gemm_nn_kernel<<<128, 256, 0, stream>>>(sim, NND, xw, h1, t);

  // ---- mean pooling over clients -> [128, 256] ----
  pool_kernel<<<128, 256, 0, stream>>>(t, out);
}